// SVBmm_74062416052398
// MI455X (gfx1250) — compile-verified
//
#include <hip/hip_runtime.h>

typedef __attribute__((ext_vector_type(2))) float v2f;
typedef __attribute__((ext_vector_type(8))) float v8f;

// Problem constants (from reference): B=4, M=16 -> BM=64 batched GEMMs
// of [QT x KVT] * [KVT x H], fp32 in / fp32 out.
#define QT  1024
#define KVT 1024
#define H   64
#define KP  128            // K-panel held in LDS
#define LDS_STRIDE 160     // floats per k-pair row: 2*H data + 32 pad (bank shift = 32)
#define PF_DIST 256        // prefetch distance along a row of s (floats, = 1KB)

// Workgroup: 256 threads = 8 waves (wave32). Each wave computes a 32x64
// output tile (two 16-row Q tiles x four 16-col N tiles) using
// V_WMMA_F32_16X16X4_F32, accumulating over K in steps of 4.
//
// LDS layout of the v panel is k-pair interleaved: element (k,h) lives at
// lds[(k>>1)*LDS_STRIDE + 2*h + (k&1)], so a B fragment {v[k][n], v[k+1][n]}
// is one contiguous ds_load_b64 into an even VGPR pair (no register movs),
// and the two lane-halves (k vs k+2 -> +LDS_STRIDE, 160 mod 64 = 32) use
// disjoint 32-bank halves -> conflict-free.
__global__ __launch_bounds__(256)
void SVBmm_74062416052398_kernel(const float* __restrict__ S,
                                 const float* __restrict__ V,
                                 float* __restrict__ O) {
    __shared__ float ldsV[(KP / 2) * LDS_STRIDE];   // 40 KB

    const int tid    = threadIdx.x;
    const int lane   = tid & 31;
    const int wave   = tid >> 5;
    const int lane15 = lane & 15;    // M / N position within a half-wave
    const int hi     = lane >> 4;    // selects K pair (A/B) or M+8 (C/D)

    const int bm     = blockIdx.y;   // 0..63
    const int qblock = blockIdx.x;   // 0..3 (256 Q rows each)

    const float* Sb = S + (size_t)bm * QT * KVT;
    const float* Vb = V + (size_t)bm * KVT * H;
    float*       Ob = O + (size_t)bm * QT * H;

    const int q0 = qblock * 256 + wave * 32;  // this wave's first Q row

    // A-fragment row pointers: lane reads s[q0 + lane%16][k + 2*(lane/16) + {0,1}]
    const float* srow0 = Sb + (size_t)(q0 + lane15) * KVT;
    const float* srow1 = srow0 + (size_t)16 * KVT;  // second Q tile

    v8f acc[2][4] = {};   // 8 accumulators of 16x16 f32 (64 VGPRs)

    for (int kp = 0; kp < KVT; kp += KP) {
        __syncthreads();  // previous panel fully consumed before overwrite

        // Cooperative load of v[kp..kp+KP) x H into the interleaved LDS
        // layout. Global side is fully coalesced float4; LDS side scatters
        // with stride 2 (worst case 2-way bank conflict, loader is tiny).
        const float* vsrc = Vb + (size_t)kp * H;
        #pragma unroll
        for (int i = 0; i < (KP * H / 4) / 256; ++i) {
            const int j  = tid + i * 256;          // float4 index
            const float4 t4 = *(const float4*)(vsrc + 4 * j);
            const int k  = j >> 4;                 // row within panel (64 f = 16 f4)
            const int h4 = (j & 15) * 4;
            float* dst = ldsV + (k >> 1) * LDS_STRIDE + (k & 1);
            dst[(h4 + 0) * 2] = t4.x;
            dst[(h4 + 1) * 2] = t4.y;
            dst[(h4 + 2) * 2] = t4.z;
            dst[(h4 + 3) * 2] = t4.w;
        }
        __syncthreads();

        for (int k2o = 0; k2o < KP; k2o += 32) {
            // Prefetch one 128B line per s row, ~1KB ahead of the consume
            // point (speculative RT prefetch; harmless at buffer tail).
            __builtin_prefetch(srow0 + kp + k2o + PF_DIST, 0, 1);
            __builtin_prefetch(srow1 + kp + k2o + PF_DIST, 0, 1);

            #pragma unroll
            for (int k2i = 0; k2i < 32; k2i += 4) {
                const int k2  = k2o + k2i;
                const int kkL = k2 + 2 * hi;   // lane's K within panel (even)
                const int kkG = kp + kkL;      // global s column

                // A fragments: streaming reads of s, non-temporal so the
                // 256MB one-pass stream doesn't evict L2-resident v.
                v2f a0 = __builtin_nontemporal_load((const v2f*)(srow0 + kkG));
                v2f a1 = __builtin_nontemporal_load((const v2f*)(srow1 + kkG));

                const float* bbase =
                    ldsV + (kkL >> 1) * LDS_STRIDE + lane15 * 2;

                #pragma unroll
                for (int nt = 0; nt < 4; ++nt) {
                    // One ds_load_b64: {v[kk][n], v[kk+1][n]} adjacent.
                    v2f b = *(const v2f*)(bbase + nt * 32);

                    acc[0][nt] = __builtin_amdgcn_wmma_f32_16x16x4_f32(
                        false, a0, false, b, (short)0, acc[0][nt], false, false);
                    acc[1][nt] = __builtin_amdgcn_wmma_f32_16x16x4_f32(
                        false, a1, false, b, (short)0, acc[1][nt], false, false);
                }
            }
        }
    }

    // Store: C/D layout -> VGPR r holds row (q0 + 16*qt + 8*hi + r), col = lane%16.
    // Output is written once and not re-read: non-temporal stores.
    #pragma unroll
    for (int qt = 0; qt < 2; ++qt) {
        #pragma unroll
        for (int nt = 0; nt < 4; ++nt) {
            #pragma unroll
            for (int r = 0; r < 8; ++r) {
                const int row = q0 + qt * 16 + 8 * hi + r;
                const int col = nt * 16 + lane15;
                __builtin_nontemporal_store(acc[qt][nt][r],
                                            Ob + (size_t)row * H + col);
            }
        }
    }
}

extern "C" void kernel_launch(void* const* d_in, const int* in_sizes, int n_in,
                              void* d_out, int out_size, void* d_ws, size_t ws_size,
                              hipStream_t stream) {
    const float* S = (const float*)d_in[0];   // [4,16,1024,1024] f32
    const float* V = (const float*)d_in[1];   // [4,16,1024,64]   f32
    float*       O = (float*)d_out;           // [4,16,1024,64]   f32

    dim3 grid(QT / 256, 64);   // 4 Q-blocks x 64 (b,m) slices
    dim3 block(256);           // 8 waves
    SVBmm_74062416052398_kernel<<<grid, block, 0, stream>>>(S, V, O);
}